// Detections_25726854103688
// MI455X (gfx1250) — compile-verified
//
#include <hip/hip_runtime.h>
#include <hip/hip_bf16.h>
#include <stdint.h>

// ---------------- problem constants ----------------
#define NBATCH 16
#define NANCH  33600
#define NDIM   85
#define NCLS   80
#define TOPK   512
#define CONF_T 0.7f
#define NMS_T  0.45f
#define MAXC   4096.0f

typedef __attribute__((ext_vector_type(2))) float v2f;
typedef __attribute__((ext_vector_type(8))) float v8f;

// order-preserving float -> uint key
__device__ __forceinline__ unsigned fkey(float f) {
    unsigned u = __float_as_uint(f);
    return u ^ ((u & 0x80000000u) ? 0xFFFFFFFFu : 0x80000000u);
}

// ---------------------------------------------------------------------------
// Kernel 1: per-anchor class max/argmax + masked score.  Memory-bound pass
// over 183 MB -> stage 128 anchors (43.5 KB) coalesced via float4 into LDS.
// ---------------------------------------------------------------------------
__global__ void __launch_bounds__(128)
score_kernel(const float* __restrict__ pred, float* __restrict__ scores,
             float* __restrict__ clsArr) {
    __shared__ float stage[128 * NDIM];
    const int t = threadIdx.x;
    const float4* gsrc = (const float4*)(pred + (size_t)blockIdx.x * 128 * NDIM);
    float4* ldst = (float4*)stage;
    for (int k = t; k < (128 * NDIM) / 4; k += 128) ldst[k] = gsrc[k];  // 2720 vec4
    __syncthreads();
    const float* p = &stage[t * NDIM];
    const float obj = p[4];
    float best = p[5];
    int bi = 0;
    #pragma unroll 4
    for (int c = 1; c < NCLS; ++c) {
        float v = p[5 + c];
        if (v > best) { best = v; bi = c; }   // first-max, matches jnp.argmax
    }
    const float s = obj * best;
    const int flat = blockIdx.x * 128 + t;
    scores[flat] = (s >= CONF_T) ? s : -1.0f;
    clsArr[flat] = (float)bi;
}

// ---------------------------------------------------------------------------
// Kernel 2: per-batch top-512 via 8-bit radix-select + 512-wide bitonic sort.
// One 1024-thread workgroup per batch; score array is L2-resident.
// ---------------------------------------------------------------------------
__global__ void __launch_bounds__(1024)
topk_kernel(const float* __restrict__ pred, const float* __restrict__ scores,
            const float* __restrict__ clsArr,
            float* __restrict__ selOff, float* __restrict__ selRaw,
            float* __restrict__ selScore, float* __restrict__ selCls) {
    const int b = blockIdx.x, t = threadIdx.x;
    __shared__ unsigned hist[256];
    __shared__ unsigned long long sel[TOPK];
    __shared__ unsigned sh_prefix, sh_krem, sh_G, sh_cntG, sh_cntE;
    const float* sc = scores + (size_t)b * NANCH;

    if (t == 0) { sh_prefix = 0u; sh_krem = TOPK; sh_G = 0u; sh_cntG = 0u; sh_cntE = 0u; }

    for (int p = 0; p < 4; ++p) {
        if (t < 256) hist[t] = 0u;
        __syncthreads();
        const unsigned prefix = sh_prefix;
        const int shift = 24 - 8 * p;
        const unsigned maskKnown = (p == 0) ? 0u : (0xFFFFFFFFu << (shift + 8));
        for (int j = t; j < NANCH; j += 1024) {
            unsigned key = fkey(sc[j]);
            if ((key & maskKnown) == prefix)
                atomicAdd(&hist[(key >> shift) & 255u], 1u);
        }
        __syncthreads();
        if (t == 0) {
            unsigned krem = sh_krem, cum = 0u;
            int v = 255;
            for (; v > 0; --v) {
                if (cum + hist[v] >= krem) break;
                cum += hist[v];
            }
            sh_prefix = prefix | ((unsigned)v << shift);
            sh_G += cum;
            sh_krem = krem - cum;
        }
        __syncthreads();
    }
    const unsigned T = sh_prefix, G = sh_G, E = sh_krem;

    // compact: strictly-greater fill [0,G), threshold-equals fill [G, 512)
    for (int j = t; j < NANCH; j += 1024) {
        unsigned key = fkey(sc[j]);
        if (key > T) {
            unsigned pos = atomicAdd(&sh_cntG, 1u);
            sel[pos] = ((unsigned long long)key << 32) | (unsigned)(~j);
        } else if (key == T) {
            unsigned e = atomicAdd(&sh_cntE, 1u);
            if (e < E) sel[G + e] = ((unsigned long long)key << 32) | (unsigned)(~j);
        }
    }
    __syncthreads();

    // bitonic sort 512 keys, descending (score desc, index asc via ~idx)
    for (int ksz = 2; ksz <= TOPK; ksz <<= 1) {
        for (int j = ksz >> 1; j > 0; j >>= 1) {
            __syncthreads();
            const int i = t;
            if (i < TOPK) {
                const int ixj = i ^ j;
                if (ixj > i) {
                    unsigned long long a = sel[i], c = sel[ixj];
                    const bool up = ((i & ksz) == 0);
                    const bool sw = up ? (a < c) : (a > c);
                    if (sw) { sel[i] = c; sel[ixj] = a; }
                }
            }
        }
    }
    __syncthreads();

    if (t < TOPK) {
        const unsigned long long sk = sel[t];
        const unsigned j = ~(unsigned)(sk & 0xFFFFFFFFull);
        const float* p5 = pred + ((size_t)b * NANCH + j) * NDIM;
        const float cx = p5[0], cy = p5[1], w = p5[2], h = p5[3];
        const float x1 = cx - 0.5f * w, y1 = cy - 0.5f * h;
        const float x2 = cx + 0.5f * w, y2 = cy + 0.5f * h;
        const float cl = clsArr[(size_t)b * NANCH + j];
        const float off = cl * MAXC;                 // CLASS_AGNOSTIC == False
        const int o4 = (b * TOPK + t) * 4;
        selOff[o4 + 0] = x1 + off; selOff[o4 + 1] = y1 + off;
        selOff[o4 + 2] = x2 + off; selOff[o4 + 3] = y2 + off;
        selRaw[o4 + 0] = x1; selRaw[o4 + 1] = y1;
        selRaw[o4 + 2] = x2; selRaw[o4 + 3] = y2;
        selScore[b * TOPK + t] = sc[j];              // == obj*conf when valid
        selCls[b * TOPK + t]   = cl;
    }
}

// ---------------------------------------------------------------------------
// Kernel 3: per-batch NMS.  512 threads (16 waves).
//  - async-to-LDS staging of boxes/scores (CDNA5 ASYNCcnt path, inline asm)
//  - area_i + area_j union term via V_WMMA_F32_16X16X4_F32 (rank-2 outer sum),
//    co-executing with the VALU min/max intersection work
//  - row boxes register-cached across all 32 column tiles (LDS traffic /8)
//  - IoU test as inter > thre*union (no division)
//  - one-wave bitmask greedy scan, then 7-column output write
// Dynamic LDS layout (base 0: no static LDS in this kernel):
//   [0      , 8192 )  boxesOff  512x4 f32
//   [8192   , 10240)  scoreS    512 f32
//   [10240  , 12288)  areaS     512 f32
//   [12288  , 14336)  keepS     512 f32
//   [14336  , 47104)  maskS     512x16 u32 (512-bit row bitmasks)
// ---------------------------------------------------------------------------
#define NMS_LDS_BYTES 47104

__global__ void __launch_bounds__(512)
nms_kernel(const float* __restrict__ selOff, const float* __restrict__ selRaw,
           const float* __restrict__ selScore, const float* __restrict__ selCls,
           float* __restrict__ out) {
    extern __shared__ char smem[];
    float*    boxesOff = (float*)smem;
    float*    scoreS   = (float*)(smem + 8192);
    float*    areaS    = (float*)(smem + 10240);
    float*    keepS    = (float*)(smem + 12288);
    unsigned* maskS    = (unsigned*)(smem + 14336);

    const int t = threadIdx.x, b = blockIdx.x;

    // ---- async stage: 512x16B boxes + 512x4B scores into LDS ----
    {
        const unsigned ldsA = (unsigned)(t * 16);
        const unsigned voff = (unsigned)(t * 16);
        const char* gb = (const char*)selOff + (size_t)b * TOPK * 16;
        asm volatile("global_load_async_to_lds_b128 %0, %1, %2"
                     :: "v"(ldsA), "v"(voff), "s"(gb) : "memory");
        if (t < 128) {
            const unsigned ldsB = 8192u + (unsigned)(t * 16);
            const char* gs = (const char*)selScore + (size_t)b * TOPK * 4;
            asm volatile("global_load_async_to_lds_b128 %0, %1, %2"
                         :: "v"(ldsB), "v"((unsigned)(t * 16)), "s"(gs) : "memory");
        }
        asm volatile("s_wait_asynccnt 0x0" ::: "memory");
    }
    __syncthreads();

    // ---- areas, clear bitmask/keep ----
    {
        const float x1 = boxesOff[t * 4 + 0], y1 = boxesOff[t * 4 + 1];
        const float x2 = boxesOff[t * 4 + 2], y2 = boxesOff[t * 4 + 3];
        areaS[t] = (x2 - x1) * (y2 - y1);
        keepS[t] = 0.f;
        #pragma unroll
        for (int w = 0; w < 16; ++w) maskS[t * 16 + w] = 0u;
    }
    __syncthreads();

    // ---- IoU > thre bitmask over 32x32 tiles of 16x16 ----
    const int wave = t >> 5, lane = t & 31;
    const bool lo = lane < 16;
    const int l16 = lane & 15;
    for (int half = 0; half < 2; ++half) {
        const int ti = wave + half * 16;                  // uniform per wave
        const int mbase = ti * 16 + (lo ? 0 : 8);
        // register-cache this lane's 8 row boxes for all 32 column tiles
        float ax1[8], ay1[8], ax2[8], ay2[8];
        #pragma unroll
        for (int r = 0; r < 8; ++r) {
            ax1[r] = boxesOff[(mbase + r) * 4 + 0];
            ay1[r] = boxesOff[(mbase + r) * 4 + 1];
            ax2[r] = boxesOff[(mbase + r) * 4 + 2];
            ay2[r] = boxesOff[(mbase + r) * 4 + 3];
        }
        // A row m: [area_i(m), 1, 0, 0]
        v2f A;
        A.x = lo ? areaS[ti * 16 + l16] : 0.f;
        A.y = lo ? 1.f : 0.f;
        for (int tj = 0; tj < 32; ++tj) {                 // uniform loop
            // B col n: [1, area_j(n), 0, 0]^T
            v2f Bm;
            Bm.x = lo ? 1.f : 0.f;
            Bm.y = lo ? areaS[tj * 16 + l16] : 0.f;
            v8f C = {};
            v8f D = __builtin_amdgcn_wmma_f32_16x16x4_f32(
                false, A, false, Bm, (short)0, C, false, false);
            // D[r]: lane<16 -> (m=r, n=lane) ; lane>=16 -> (m=r+8, n=lane-16)
            const int gj = tj * 16 + l16;
            const float bx1 = boxesOff[gj * 4 + 0], by1 = boxesOff[gj * 4 + 1];
            const float bx2 = boxesOff[gj * 4 + 2], by2 = boxesOff[gj * 4 + 3];
            #pragma unroll
            for (int r = 0; r < 8; ++r) {
                const int gi = mbase + r;
                float iw = fminf(ax2[r], bx2) - fmaxf(ax1[r], bx1);
                float ih = fminf(ay2[r], by2) - fmaxf(ay1[r], by1);
                iw = fmaxf(iw, 0.f); ih = fmaxf(ih, 0.f);
                const float inter = iw * ih;
                const float uni = fmaxf(D[r] - inter, 1e-9f); // a_i+a_j-inter
                // inter/uni > thre  <=>  inter > thre*uni   (uni > 0)
                const bool bit = (inter > NMS_T * uni) && (gi != gj);
                if (bit) atomicOr(&maskS[gi * 16 + (gj >> 5)], 1u << (gj & 31));
            }
        }
    }
    __syncthreads();

    // ---- greedy scan: one wave, 512-bit suppressed mask in lanes 0..15 ----
    if (t < 32) {
        unsigned supp = 0u;
        for (int i = 0; i < TOPK; ++i) {
            const unsigned sw = __shfl(supp, i >> 5, 32);
            const bool keep_i = (scoreS[i] >= CONF_T) && (((sw >> (i & 31)) & 1u) == 0u);
            if (keep_i && t < 16) supp |= maskS[i * 16 + t];
            if (t == 0) keepS[i] = keep_i ? 1.f : 0.f;
        }
    }
    __syncthreads();

    // ---- output: [x1,y1,x2,y2, obj*conf, class, keep] * keep ----
    {
        const float kf = keepS[t];
        const int idx = b * TOPK + t;
        const float* rb = selRaw + idx * 4;
        float* o = out + (size_t)idx * 7;
        o[0] = rb[0] * kf; o[1] = rb[1] * kf;
        o[2] = rb[2] * kf; o[3] = rb[3] * kf;
        o[4] = selScore[idx] * kf;
        o[5] = selCls[idx] * kf;
        o[6] = kf;
    }
}

// ---------------------------------------------------------------------------
// Kernel 4: targets_out  (16x50x5 -> 16x50x6)
// ---------------------------------------------------------------------------
__global__ void targets_kernel(const float* __restrict__ tg, float* __restrict__ out2) {
    const int i = blockIdx.x * blockDim.x + threadIdx.x;
    if (i >= NBATCH * 50) return;
    const float* p = tg + i * 5;
    const float lab = p[0], cx = p[1], cy = p[2], w = p[3], h = p[4];
    const float v = (lab > 0.f) ? 1.f : 0.f;
    float* o = out2 + i * 6;
    o[0] = (cx - 0.5f * w) * v; o[1] = (cy - 0.5f * h) * v;
    o[2] = (cx + 0.5f * w) * v; o[3] = (cy + 0.5f * h) * v;
    o[4] = lab * v; o[5] = v;
}

// ---------------------------------------------------------------------------
extern "C" void kernel_launch(void* const* d_in, const int* in_sizes, int n_in,
                              void* d_out, int out_size, void* d_ws, size_t ws_size,
                              hipStream_t stream) {
    (void)in_sizes; (void)n_in; (void)out_size; (void)ws_size;
    const float* pred = (const float*)d_in[0];   // (16, 33600, 85)
    const float* targ = (const float*)d_in[1];   // (16, 50, 5)
    float* out = (float*)d_out;                  // 16*512*7 + 16*50*6

    // workspace layout (floats), all regions fully rewritten every call
    float* scores   = (float*)d_ws;                       // 16*33600
    float* clsArr   = scores   + NBATCH * NANCH;          // 16*33600
    float* selOff   = clsArr   + NBATCH * NANCH;          // 16*512*4
    float* selRaw   = selOff   + NBATCH * TOPK * 4;       // 16*512*4
    float* selScore = selRaw   + NBATCH * TOPK * 4;       // 16*512
    float* selCls   = selScore + NBATCH * TOPK;           // 16*512

    const int nblk1 = (NBATCH * NANCH) / 128;             // 4200, exact
    score_kernel<<<nblk1, 128, 0, stream>>>(pred, scores, clsArr);
    topk_kernel<<<NBATCH, 1024, 0, stream>>>(pred, scores, clsArr,
                                             selOff, selRaw, selScore, selCls);
    nms_kernel<<<NBATCH, 512, NMS_LDS_BYTES, stream>>>(selOff, selRaw,
                                                       selScore, selCls, out);
    targets_kernel<<<(NBATCH * 50 + 255) / 256, 256, 0, stream>>>(
        targ, out + NBATCH * TOPK * 7);
}